// Outer_Product_Mean_8538394985035
// MI455X (gfx1250) — compile-verified
//
#include <hip/hip_runtime.h>
#include <hip/hip_bf16.h>

typedef __attribute__((ext_vector_type(2))) float v2f;
typedef __attribute__((ext_vector_type(8))) float v8f;

#define S_DIM  128
#define R_DIM  256
#define CM_DIM 256
#define C_DIM  32
#define CZ_DIM 128
#define RC     (R_DIM * C_DIM)   /* 8192 columns of the activation matrix */
#define LDS_STRIDE 1028          /* 1024 + 4 floats pad: 16-row A-frag reads hit 16 banks */

// Swizzled layouts (K-pair interleaved so a WMMA fragment = one b64 load/lane):
//   a   : element (k=s, col)  at  a  [((k>>1)*RC  + col)*2 + (k&1)]
//   w2s : element (k,   z  )  at  w2s[ (k>>1)*256 + z*2   + (k&1)]

__device__ __forceinline__ v8f wmma4(v2f A, v2f B, v8f C) {
  // D(16x16,f32) = A(16x4,f32) * B(4x16,f32) + C   — exact fp32 matrix op
  return __builtin_amdgcn_wmma_f32_16x16x4_f32(false, A, false, B, (short)0, C,
                                               false, false);
}

// ---------------------------------------------------------------------------
// Kernel 0: one-time swizzle of w2 (1024x128) into K-pair-interleaved layout.
// ---------------------------------------------------------------------------
__global__ void __launch_bounds__(256)
opm_swz_w2_kernel(const float* __restrict__ w2, float* __restrict__ w2s) {
  int idx = blockIdx.x * 256 + threadIdx.x;       // 0 .. 131071 (output index)
  int kp  = idx >> 8;                             // k pair, 0..511
  int rem = idx & 255;
  int z   = rem >> 1;
  int par = rem & 1;
  w2s[idx] = w2[(size_t)(kp * 2 + par) * CZ_DIM + z];
}

// ---------------------------------------------------------------------------
// Kernel 1: LayerNorm over CM + fc1 (CM->C).  One wave32 per (s,r) row.
// Writes the activation in the K-pair-interleaved layout above.
// ---------------------------------------------------------------------------
__global__ void __launch_bounds__(256)
opm_ln_fc1_kernel(const float* __restrict__ x, const float* __restrict__ gamma,
                  const float* __restrict__ beta, const float* __restrict__ w1,
                  const float* __restrict__ b1, float* __restrict__ a) {
  __shared__ float xn[8][CM_DIM];
  const int tid  = threadIdx.x;
  const int wave = tid >> 5;
  const int lane = tid & 31;
  const int row  = blockIdx.x * 8 + wave;          // (s*R + r), 0..32767
  const float* xr = x + (size_t)row * CM_DIM;

  float v[8];
  float s = 0.f;
#pragma unroll
  for (int m = 0; m < 8; ++m) { v[m] = xr[lane + m * 32]; s += v[m]; }
#pragma unroll
  for (int off = 16; off > 0; off >>= 1) s += __shfl_xor(s, off, 32);
  const float mu = s * (1.0f / 256.0f);

  float q = 0.f;
#pragma unroll
  for (int m = 0; m < 8; ++m) { float d = v[m] - mu; q += d * d; }
#pragma unroll
  for (int off = 16; off > 0; off >>= 1) q += __shfl_xor(q, off, 32);
  const float rstd = rsqrtf(q * (1.0f / 256.0f) + 1e-5f);

#pragma unroll
  for (int m = 0; m < 8; ++m) {
    int idx = lane + m * 32;
    xn[wave][idx] = (v[m] - mu) * rstd * gamma[idx] + beta[idx];
  }
  __syncthreads();

  // fc1: lane == output channel c; xn[wave][m] is an LDS broadcast read.
  float acc = b1[lane];
#pragma unroll 4
  for (int m = 0; m < CM_DIM; ++m)
    acc = fmaf(xn[wave][m], w1[m * C_DIM + lane], acc);

  const int s_idx = row >> 8;                      // s
  const int r_idx = row & 255;                     // r
  a[((size_t)(s_idx >> 1) * RC + r_idx * C_DIM + lane) * 2 + (s_idx & 1)] = acc;
}

// ---------------------------------------------------------------------------
// Kernel 2: fused  gram (op[i,j,c,d] = (1/S) a_i^T a_j)  +  fc2 (1024 -> 128).
// Block = (i, 16-wide j-block).  8 waves.  op slab lives only in LDS (64 KB).
//   gram : M=32 (c), N=512 (j_local*32+d), K=128 (s)  -> 2048 V_WMMA f32
//          6 x global_load_b64 feed 8 WMMAs per K-step (2 A-frag x 4 B-frag)
//   fc2  : M=16 (pairs), K=1024 (c*32+d), N=128 (z)   -> 2048 V_WMMA f32
//          1 x ds_load_b64 + 1 x global_load_b64 per WMMA
// ---------------------------------------------------------------------------
__global__ void __launch_bounds__(256)
opm_gram_fc2_kernel(const float* __restrict__ a, const float* __restrict__ w2s,
                    const float* __restrict__ b2, float* __restrict__ out) {
  __shared__ float G[16 * LDS_STRIDE];             // 65,792 B
  const int jb   = blockIdx.x;                     // 0..15  (j block)
  const int i    = blockIdx.y;                     // 0..255
  const int tid  = threadIdx.x;
  const int wave = tid >> 5;
  const int lane = tid & 31;
  const int half = lane >> 4;                      // 0/1: K-half of the frag
  const int l16  = lane & 15;

  const int colA0 = i * C_DIM + l16;               // a_i columns  (+ ct*16)
  const int colB0 = jb * 512 + wave * 64 + l16;    // a_J columns  (+ t*16)

  v8f acc[8];
  const v8f vzero = {0.f, 0.f, 0.f, 0.f, 0.f, 0.f, 0.f, 0.f};
#pragma unroll
  for (int qn = 0; qn < 8; ++qn) acc[qn] = vzero;

#pragma unroll 1
  for (int k0 = 0; k0 < S_DIM; k0 += 4) {
    // k-pair row for this lane: holds elements {k, k+1} interleaved per col
    const float* base = a + (size_t)((k0 >> 1) + half) * (RC * 2);
    v2f Af[2], Bf[4];
#pragma unroll
    for (int ct = 0; ct < 2; ++ct)                 // A-frags (c tiles)
      Af[ct] = *(const v2f*)(base + (size_t)(colA0 + ct * 16) * 2);
#pragma unroll
    for (int t = 0; t < 4; ++t)                    // B-frags (jd tiles)
      Bf[t] = *(const v2f*)(base + (size_t)(colB0 + t * 16) * 2);
#pragma unroll
    for (int ct = 0; ct < 2; ++ct)
#pragma unroll
      for (int t = 0; t < 4; ++t)
        acc[ct * 4 + t] = wmma4(Af[ct], Bf[t], acc[ct * 4 + t]);
  }

  // Scatter D-frags into the LDS op slab: G[pair][c*32+d], scaled by 1/S.
  const float sinv = 1.0f / (float)S_DIM;
#pragma unroll
  for (int ct = 0; ct < 2; ++ct) {
#pragma unroll
    for (int t = 0; t < 4; ++t) {
#pragma unroll
      for (int r = 0; r < 8; ++r) {
        int c    = ct * 16 + r + half * 8;         // row of D tile -> c index
        int colg = wave * 64 + t * 16 + l16;       // 0..511 = j_local*32 + d
        int jl   = colg >> 5;
        int d    = colg & 31;
        G[jl * LDS_STRIDE + c * C_DIM + d] = acc[ct * 4 + t][r] * sinv;
      }
    }
  }
  __syncthreads();

  // fc2: each wave owns one 16-wide z tile.
  const int z0 = wave * 16;
  v8f o = vzero;
#pragma unroll 1
  for (int k0 = 0; k0 < C_DIM * C_DIM; k0 += 4) {
    v2f Afr = *(const v2f*)(&G[l16 * LDS_STRIDE + k0 + half * 2]);  // ds_load_b64
    v2f Bfr = *(const v2f*)(w2s + (size_t)((k0 >> 1) + half) * 256
                                 + (z0 + l16) * 2);                 // global b64
    o = wmma4(Afr, Bfr, o);
  }

  const float bz = b2[z0 + l16];
#pragma unroll
  for (int r = 0; r < 8; ++r) {
    int pair = r + half * 8;
    int j    = jb * 16 + pair;
    out[((size_t)i * R_DIM + j) * CZ_DIM + z0 + l16] = o[r] + bz;
  }
}

// ---------------------------------------------------------------------------
extern "C" void kernel_launch(void* const* d_in, const int* in_sizes, int n_in,
                              void* d_out, int out_size, void* d_ws, size_t ws_size,
                              hipStream_t stream) {
  const float* x     = (const float*)d_in[0];
  const float* gamma = (const float*)d_in[1];
  const float* beta  = (const float*)d_in[2];
  const float* w1    = (const float*)d_in[3];
  const float* b1    = (const float*)d_in[4];
  const float* w2    = (const float*)d_in[5];
  const float* b2    = (const float*)d_in[6];
  float* out = (float*)d_out;

  float* a   = (float*)d_ws;                        // S*R*C floats   = 4 MB
  float* w2s = a + (size_t)S_DIM * RC;              // 1024*128 floats = 512 KB

  opm_swz_w2_kernel<<<(C_DIM * C_DIM * CZ_DIM) / 256, 256, 0, stream>>>(w2, w2s);
  opm_ln_fc1_kernel<<<(S_DIM * R_DIM) / 8, 256, 0, stream>>>(x, gamma, beta, w1, b1, a);
  opm_gram_fc2_kernel<<<dim3(R_DIM / 16, R_DIM), 256, 0, stream>>>(a, w2s, b2, out);
}